// CTCLoss_36541581754866
// MI455X (gfx1250) — compile-verified
//
#include <hip/hip_runtime.h>

// Problem constants (match the reference)
#define BB 32
#define TT 1024
#define VV 1024
#define LL 256
#define SS (2 * LL + 1)      // 513 extended states
#define NEGI (-1e30f)        // finite log-zero
#define NTHREADS 544         // 17 wave32s; covers S=513 one state per lane
#define PIPE 4               // async rows in flight
#define NBUF (PIPE + 1)      // row buffers (one extra so issue never clobbers live row)

// ---- CDNA5 async global->LDS staging (guarded; sync fallback otherwise) ----
#if __has_builtin(__builtin_amdgcn_global_load_async_to_lds_b128)
#define HAVE_ASYNC_LDS 1
typedef int v4i __attribute__((vector_size(16)));
typedef __attribute__((address_space(1))) v4i* gv4p;  // global src (non-const)
typedef __attribute__((address_space(3))) v4i* lv4p;  // LDS dst
#else
#define HAVE_ASYNC_LDS 0
#endif

// Wait until at most N async global->LDS ops remain outstanding (per wave).
#define ASYNC_WAIT(N)                                        \
  do {                                                       \
    if (HAVE_ASYNC_LDS) {                                    \
      __builtin_amdgcn_s_wait_asynccnt(N);                   \
    }                                                        \
  } while (0)

// Copy one 4KB row (VV floats) global -> LDS. Threads tid < VV/4 move 16B each.
__device__ __forceinline__ void stage_row(const float* __restrict__ gsrc,
                                          float* ldst, int tid) {
  if (tid < VV / 4) {
#if HAVE_ASYNC_LDS
    __builtin_amdgcn_global_load_async_to_lds_b128(
        (gv4p)(const_cast<float*>(gsrc) + tid * 4),
        (lv4p)(ldst + tid * 4), 0, 0);
#else
    float4 v = *(const float4*)(gsrc + tid * 4);
    *(float4*)(ldst + tid * 4) = v;
#endif
  }
}

__global__ __launch_bounds__(NTHREADS) void ctc_forward_kernel(
    const float* __restrict__ log_probs,   // (B,T,V)
    const int* __restrict__ targets,       // (B,L)
    const int* __restrict__ input_lengths, // (B,)
    const int* __restrict__ target_lengths,// (B,)
    float* __restrict__ out_per_b) {       // (B,) loss_b / target_len_b
  const int b = blockIdx.x;
  const int tid = threadIdx.x;
  const int s = tid;  // extended-state index handled by this lane
  const float* lpb = log_probs + (size_t)b * TT * VV;

  // NBUF-deep 4KB probability rows + double-buffered alpha with 2 pad slots.
  __shared__ float rowbuf[NBUF][VV];
  __shared__ float abuf[2][SS + 2];

  // Per-lane extended label + skip-transition predicate (computed once).
  int e = 0;          // blank for even s
  bool can_skip = false;
  if (s < SS) {
    if (s & 1) {
      e = targets[b * LL + (s >> 1)];
      int em2 = (s >= 3) ? targets[b * LL + ((s - 2) >> 1)] : -1;
      can_skip = (e != em2);  // labels are in [1,V) so e != blank always
    }
  }

  const int Tb = input_lengths[b];  // uniform across the block

  // Init alpha(t=0) and the NEG pad slots.
  if (tid < 2) {
    abuf[0][tid] = NEGI;
    abuf[1][tid] = NEGI;
  }
  if (s < SS) {
    float a0 = NEGI;
    if (s == 0) a0 = lpb[0];       // log p(blank) at t=0
    else if (s == 1) a0 = lpb[e];  // log p(y1)    at t=0
    abuf[0][2 + s] = a0;
  }

  // Prologue: issue rows 1..PIPE (row k -> buffer k % NBUF).
  {
    int kmax = Tb - 1;
    if (kmax > PIPE) kmax = PIPE;
    for (int k = 1; k <= kmax; ++k)
      stage_row(lpb + (size_t)k * VV, rowbuf[k % NBUF], tid);
  }

  int cur = 0;        // abuf[cur] holds alpha_{t-1}
  int bc = 1 % NBUF;  // buffer holding row t (t starts at 1)
  int bi = (1 + PIPE) % NBUF;  // buffer for next issued row (t+PIPE)
  for (int t = 1; t < Tb; ++t) {
    // Steady state: PIPE rows outstanding; oldest (row t) done once cnt<=PIPE-1.
    // Tail (last PIPE-1 steps): fewer outstanding; wait for all.
    if (t <= Tb - PIPE) ASYNC_WAIT(PIPE - 1);
    else                ASYNC_WAIT(0);
    __syncthreads();  // alpha_{t-1} + row t visible; row t-1 fully consumed

    // Issue row t+PIPE into row t-1's buffer (just freed by the barrier).
    if (t + PIPE < Tb) stage_row(lpb + (size_t)(t + PIPE) * VV, rowbuf[bi], tid);

    if (s < SS) {
      const float* ap = abuf[cur];
      float a  = ap[2 + s];
      float a1 = ap[1 + s];
      float a2 = can_skip ? ap[s] : NEGI;
      float m = fmaxf(a, fmaxf(a1, a2));
      float sum = __expf(a - m) + __expf(a1 - m) + __expf(a2 - m);
      float nv = m + __logf(sum) + rowbuf[bc][e];
      abuf[cur ^ 1][2 + s] = nv;
    }
    cur ^= 1;
    if (++bc == NBUF) bc = 0;
    if (++bi == NBUF) bi = 0;
    // write->read hazard on abuf is covered by the barrier at loop top
  }
  __syncthreads();

  if (tid == 0) {
    int tl = target_lengths[b];
    int end = 2 * tl;  // last blank state; end-1 = last label state
    float aE = abuf[cur][2 + end];
    float aP = abuf[cur][1 + end];
    float m = fmaxf(aE, aP);
    float ll = m + __logf(__expf(aE - m) + __expf(aP - m));
    float loss = -ll;
    if (!(loss < 1e29f)) loss = 0.0f;  // zero_infinity (also kills NaN)
    out_per_b[b] = loss / (float)tl;
  }
}

__global__ void ctc_reduce_kernel(const float* __restrict__ per_b,
                                  float* __restrict__ out) {
  if (threadIdx.x == 0) {
    float acc = 0.0f;
    for (int i = 0; i < BB; ++i) acc += per_b[i];
    out[0] = acc / (float)BB;
  }
}

extern "C" void kernel_launch(void* const* d_in, const int* in_sizes, int n_in,
                              void* d_out, int out_size, void* d_ws, size_t ws_size,
                              hipStream_t stream) {
  const float* log_probs = (const float*)d_in[0];
  const int* targets = (const int*)d_in[1];
  const int* input_lengths = (const int*)d_in[2];
  const int* target_lengths = (const int*)d_in[3];
  float* per_b = (float*)d_ws;  // 32 floats of scratch

  ctc_forward_kernel<<<BB, NTHREADS, 0, stream>>>(
      log_probs, targets, input_lengths, target_lengths, per_b);
  ctc_reduce_kernel<<<1, 32, 0, stream>>>(per_b, (float*)d_out);
}